// Net_31576599560702
// MI455X (gfx1250) — compile-verified
//
#include <hip/hip_runtime.h>

typedef __attribute__((ext_vector_type(2))) float v2f;
typedef __attribute__((ext_vector_type(8))) float v8f;

__device__ __forceinline__ v8f wmma4(v2f a, v2f b, v8f c) {
  // V_WMMA_F32_16X16X4_F32 : full fp32 matmul on the matrix pipe
  return __builtin_amdgcn_wmma_f32_16x16x4_f32(false, a, false, b, (short)0, c,
                                               false, false);
}

// ---- CDNA5 async global->LDS staging (ASYNCcnt path), guarded ------------
#if defined(__has_builtin)
#if __has_builtin(__builtin_amdgcn_global_load_async_to_lds_b32)
#define HAS_ASYNC_LDS 1
#endif
#if __has_builtin(__builtin_amdgcn_s_wait_asynccnt)
#define HAS_WAIT_ASYNC 1
#endif
#endif

typedef __attribute__((address_space(1))) int as1_int;
typedef __attribute__((address_space(3))) int as3_int;

__device__ __forceinline__ void async_g2l_b32(const float* gsrc, float* ldst) {
#ifdef HAS_ASYNC_LDS
  // generic->AS1 and generic->AS3 via integer cast (LDS offset = addr[31:0])
  __builtin_amdgcn_global_load_async_to_lds_b32(
      (as1_int*)(size_t)gsrc,
      (as3_int*)(size_t)(unsigned int)(size_t)ldst, 0, 0);
#else
  *ldst = *gsrc;
#endif
}

__device__ __forceinline__ void wait_async_all() {
#ifdef HAS_ASYNC_LDS
#ifdef HAS_WAIT_ASYNC
  __builtin_amdgcn_s_wait_asynccnt(0);
#else
  asm volatile("s_wait_asynccnt 0" ::: "memory");
#endif
#endif
}

// ---------------------------------------------------------------------------
// Kernel 1: per-sample GCN (2 graph convs, all 4 GEMMs on WMMA) -> hg (B,20)
// One block per batch sample, 256 threads (8 wave32).
// ---------------------------------------------------------------------------
__global__ __launch_bounds__(256) void gcn_kernel(
    const float* __restrict__ adj, const float* __restrict__ feat,
    const float* __restrict__ Wg1, const float* __restrict__ Wg2,
    float* __restrict__ hg) {
  extern __shared__ float smem[];
  float* sA  = smem;              // 64 x 65   adjacency
  float* sB  = sA + 64 * 65;      // 64 x 116  feat, then H1*d (padded 112)
  float* sS  = sB + 64 * 116;     // 64 x 116  S1, then S2
  float* sW1 = sS + 64 * 116;     // 64 x 116  W_gc1 (cols 100..111 zero)
  float* sW2 = sW1 + 64 * 116;    // 100 x 33  W_gc2 (cols 20..31 zero)
  float* sD  = sW2 + 100 * 33;    // 64        deg^-1/2
  float* sH2 = sD + 64;           // 64 x 20

  const int b = blockIdx.x;
  const int tid = threadIdx.x;
  const int lane = tid & 31, wave = tid >> 5;
  const int half = lane >> 4, l = lane & 15;

  const float* Ab = adj + (size_t)b * 4096;
  const float* Xb = feat + (size_t)b * 4096;

  // ---- issue async DMA: feat, W1, W2 stream into LDS in the background ----
  for (int i = tid; i < 4096; i += 256) {
    int n = i >> 6, k = i & 63;
    async_g2l_b32(Xb + i, &sB[n * 116 + k]);
  }
  for (int i = tid; i < 6400; i += 256) {
    int k = i / 100, c = i - k * 100;
    async_g2l_b32(Wg1 + i, &sW1[k * 116 + c]);
  }
  for (int i = tid; i < 2000; i += 256) {
    int k = i / 20, c = i - k * 20;
    async_g2l_b32(Wg2 + i, &sW2[k * 33 + c]);
  }
  // zero pads (ds stores, disjoint from async targets)
  for (int i = tid; i < 64 * 12; i += 256)
    sW1[(i / 12) * 116 + 100 + (i % 12)] = 0.f;
  for (int i = tid; i < 100 * 12; i += 256)
    sW2[(i / 12) * 33 + 20 + (i % 12)] = 0.f;

  // ---- load A through VGPRs meanwhile ----
  for (int i = tid; i < 4096; i += 256) {
    int n = i >> 6, k = i & 63;
    sA[n * 65 + k] = Ab[i];
  }
  __syncthreads();  // sA + pads visible (async may still be in flight)

  // degree, d = clip(deg,eps)^-1/2 (overlaps async streaming)
  if (tid < 64) {
    float s = 0.f;
    const float* row = &sA[tid * 65];
    for (int m = 0; m < 64; ++m) s += row[m];
    sD[tid] = 1.0f / sqrtf(fmaxf(s, 1e-12f));
  }
  wait_async_all();
  __syncthreads();  // sB/sW1/sW2/sD all visible

  // GEMM1 (WMMA): S1 = A @ (X*d), d folded into B fragments. 16 tiles.
  for (int t = wave; t < 16; t += 8) {
    int m0 = (t >> 2) * 16, n0 = (t & 3) * 16;
    v8f acc = {};
    const float* Arow = &sA[(m0 + l) * 65 + 2 * half];
#pragma unroll
    for (int k0 = 0; k0 < 64; k0 += 4) {
      float s0 = sD[k0 + 2 * half], s1 = sD[k0 + 2 * half + 1];
      v2f av, bv;
      av[0] = Arow[k0];
      av[1] = Arow[k0 + 1];
      bv[0] = sB[(k0 + 2 * half) * 116 + n0 + l] * s0;
      bv[1] = sB[(k0 + 2 * half + 1) * 116 + n0 + l] * s1;
      acc = wmma4(av, bv, acc);
    }
#pragma unroll
    for (int v = 0; v < 8; ++v)
      sS[(m0 + v + 8 * half) * 116 + n0 + l] = acc[v];
  }
  __syncthreads();

  // H1-GEMM (WMMA): H1 = relu((S1*d) @ W1), store H1*d. M=64,N=112,K=64.
  for (int t = wave; t < 28; t += 8) {
    int m0 = (t / 7) * 16, n0 = (t % 7) * 16;
    v8f acc = {};
    const float dm = sD[m0 + l];
    const float* Arow = &sS[(m0 + l) * 116 + 2 * half];
#pragma unroll
    for (int k0 = 0; k0 < 64; k0 += 4) {
      v2f av, bv;
      av[0] = Arow[k0] * dm;
      av[1] = Arow[k0 + 1] * dm;
      bv[0] = sW1[(k0 + 2 * half) * 116 + n0 + l];
      bv[1] = sW1[(k0 + 2 * half + 1) * 116 + n0 + l];
      acc = wmma4(av, bv, acc);
    }
#pragma unroll
    for (int v = 0; v < 8; ++v) {
      int row = m0 + v + 8 * half;
      sB[row * 116 + n0 + l] = fmaxf(acc[v], 0.f) * sD[row];
    }
  }
  __syncthreads();

  // GEMM2 (WMMA): S2 = A @ (H1*d). M=64, N=112, K=64.
  for (int t = wave; t < 28; t += 8) {
    int m0 = (t / 7) * 16, n0 = (t % 7) * 16;
    v8f acc = {};
    const float* Arow = &sA[(m0 + l) * 65 + 2 * half];
#pragma unroll
    for (int k0 = 0; k0 < 64; k0 += 4) {
      v2f av, bv;
      av[0] = Arow[k0];
      av[1] = Arow[k0 + 1];
      bv[0] = sB[(k0 + 2 * half) * 116 + n0 + l];
      bv[1] = sB[(k0 + 2 * half + 1) * 116 + n0 + l];
      acc = wmma4(av, bv, acc);
    }
#pragma unroll
    for (int v = 0; v < 8; ++v)
      sS[(m0 + v + 8 * half) * 116 + n0 + l] = acc[v];
  }
  __syncthreads();

  // H2-GEMM (WMMA): H2 = relu((S2*d) @ W2). M=64, N=32(pad), K=100. 8 tiles.
  {
    int m0 = (wave >> 1) * 16, n0 = (wave & 1) * 16;
    v8f acc = {};
    const float dm = sD[m0 + l];
    const float* Arow = &sS[(m0 + l) * 116 + 2 * half];
#pragma unroll
    for (int k0 = 0; k0 < 100; k0 += 4) {
      v2f av, bv;
      av[0] = Arow[k0] * dm;
      av[1] = Arow[k0 + 1] * dm;
      bv[0] = sW2[(k0 + 2 * half) * 33 + n0 + l];
      bv[1] = sW2[(k0 + 2 * half + 1) * 33 + n0 + l];
      acc = wmma4(av, bv, acc);
    }
    if (n0 + l < 20) {
#pragma unroll
      for (int v = 0; v < 8; ++v)
        sH2[(m0 + v + 8 * half) * 20 + n0 + l] = fmaxf(acc[v], 0.f);
    }
  }
  __syncthreads();

  // hg = mean over nodes
  if (tid < 20) {
    float s = 0.f;
    for (int n = 0; n < 64; ++n) s += sH2[n * 20 + tid];
    hg[(size_t)b * 20 + tid] = s * (1.0f / 64.0f);
  }
}

// ---------------------------------------------------------------------------
// Kernel 2: tokenizer + bilinear attention (2 glimpses) + FC head, fused.
// One block per batch sample, 256 threads (8 wave32). M padded 200 -> 208.
// ---------------------------------------------------------------------------
__global__ __launch_bounds__(256) void attn_head_kernel(
    const float* __restrict__ desc, const float* __restrict__ id_emb,
    const float* __restrict__ val_w1, const float* __restrict__ val_b1,
    const float* __restrict__ val_w2, const float* __restrict__ val_b2,
    const float* __restrict__ tok_g, const float* __restrict__ tok_b,
    const float* __restrict__ U_attn, const float* __restrict__ V_attn,
    const float* __restrict__ p_g, const float* __restrict__ U_joint,
    const float* __restrict__ V_joint, const float* __restrict__ P,
    const float* __restrict__ res_g, const float* __restrict__ res_b,
    const float* __restrict__ fc1_w, const float* __restrict__ fc1_b,
    const float* __restrict__ fc2_w, const float* __restrict__ fc2_b,
    const float* __restrict__ fc3_w, const float* __restrict__ fc3_b,
    const float* __restrict__ bn1_g, const float* __restrict__ bn1_b,
    const float* __restrict__ bn2_g, const float* __restrict__ bn2_b,
    const float* __restrict__ hg, float* __restrict__ out,
    float* __restrict__ attn0, float* __restrict__ attn1) {
  extern __shared__ float smem[];
  float* sV1  = smem;                 // 208 x 33  val1 (rows 200..207 zero)
  float* sY   = sV1 + 208 * 33;       // 208 x 65  tmp/Y
  float* sW   = sY + 208 * 65;        // 2048      w2 (32 x 64)
  float* sWv  = sW + 2048;            // 2048      [V_attn|V_joint] (64 x 32)
  float* sYV  = sWv + 2048;           // 208 x 33  cols 0..15 attn, 16..31 joint
  float* sRed = sYV + 208 * 33;       // 256 reduction buffer
  float* sVec = sRed + 256;           // 256 attention weights
  float* sO1  = sVec + 256;           // 256
  float* sF   = sO1 + 256;            // 20
  float* sXU  = sF + 20;              // 16
  float* sVy  = sXU + 16;             // 16
  float* sXV  = sVy + 16;             // 16
  float* sLN  = sXV + 16;             // 8

  const int b = blockIdx.x;
  const int tid = threadIdx.x;
  const int lane = tid & 31, wave = tid >> 5;
  const int half = lane >> 4, l = lane & 15;

  if (tid == 0) { __builtin_prefetch(fc1_w, 0, 3); __builtin_prefetch(fc2_w, 0, 3); }

  // ---- async DMA: w2 and interleaved [V_attn|V_joint] stream into LDS ----
  for (int i = tid; i < 2048; i += 256) async_g2l_b32(val_w2 + i, &sW[i]);
  for (int i = tid; i < 2048; i += 256) {
    int k = i >> 5, j = i & 31;
    const float* src = (j < 16) ? (V_attn + k * 16 + j)
                                : (V_joint + k * 16 + (j - 16));
    async_g2l_b32(src, &sWv[i]);
  }

  // val1 = relu(v*w1 + b1) (rank-1 outer product), rows 200..207 zero
  for (int i = tid; i < 208 * 32; i += 256) {
    int m = i >> 5, j = i & 31;
    float v1 = 0.f;
    if (m < 200) {
      float v = desc[(size_t)b * 200 + m];
      v1 = fmaxf(fmaf(v, val_w1[j], val_b1[j]), 0.f);
    }
    sV1[m * 33 + j] = v1;
  }
  wait_async_all();
  __syncthreads();

  // GEMM (WMMA): tmpY = val1 @ w2, M=208, N=64, K=32 -> 52 tiles
  for (int t = wave; t < 52; t += 8) {
    int m0 = (t >> 2) * 16, n0 = (t & 3) * 16;
    v8f acc = {};
    const float* Arow = &sV1[(m0 + l) * 33 + 2 * half];
#pragma unroll
    for (int k0 = 0; k0 < 32; k0 += 4) {
      v2f av, bv;
      av[0] = Arow[k0];
      av[1] = Arow[k0 + 1];
      bv[0] = sW[(k0 + 2 * half) * 64 + n0 + l];
      bv[1] = sW[(k0 + 2 * half + 1) * 64 + n0 + l];
      acc = wmma4(av, bv, acc);
    }
#pragma unroll
    for (int v = 0; v < 8; ++v)
      sY[(m0 + v + 8 * half) * 65 + n0 + l] = acc[v];
  }
  __syncthreads();

  // LayerNorm rows -> Y (tok_g/tok_b), pad rows 200..207 with zeros
  if (tid < 200) {
    float* row = &sY[tid * 65];
    const float* em = &id_emb[tid * 64];
    float mu = 0.f;
    for (int k = 0; k < 64; ++k) {
      float x = row[k] + val_b2[k] + em[k];
      row[k] = x;
      mu += x;
    }
    mu *= (1.f / 64.f);
    float var = 0.f;
    for (int k = 0; k < 64; ++k) { float d0 = row[k] - mu; var = fmaf(d0, d0, var); }
    var *= (1.f / 64.f);
    float rs = 1.0f / sqrtf(var + 1e-5f);
    for (int k = 0; k < 64; ++k)
      row[k] = fmaf((row[k] - mu) * rs, tok_g[k], tok_b[k]);
  } else if (tid < 208) {
    float* row = &sY[tid * 65];
    for (int k = 0; k < 64; ++k) row[k] = 0.f;
  }
  __syncthreads();

  // GEMM (WMMA): YV = relu(Y @ [V_attn|V_joint]), M=208, N=32, K=64
  for (int t = wave; t < 26; t += 8) {
    int m0 = (t >> 1) * 16, n0 = (t & 1) * 16;
    v8f acc = {};
    const float* Arow = &sY[(m0 + l) * 65 + 2 * half];
#pragma unroll
    for (int k0 = 0; k0 < 64; k0 += 4) {
      v2f av, bv;
      av[0] = Arow[k0];
      av[1] = Arow[k0 + 1];
      bv[0] = sWv[(k0 + 2 * half) * 32 + n0 + l];
      bv[1] = sWv[(k0 + 2 * half + 1) * 32 + n0 + l];
      acc = wmma4(av, bv, acc);
    }
#pragma unroll
    for (int v = 0; v < 8; ++v)
      sYV[(m0 + v + 8 * half) * 33 + n0 + l] = fmaxf(acc[v], 0.f);
  }
  if (tid < 20) sF[tid] = hg[(size_t)b * 20 + tid];
  __syncthreads();

  // Glimpse loop
  for (int g = 0; g < 2; ++g) {
    if (tid < 16) {
      float a0 = 0.f, a1 = 0.f;
      for (int j = 0; j < 20; ++j) {
        a0 = fmaf(sF[j], U_attn[j * 16 + tid], a0);
        a1 = fmaf(sF[j], U_joint[j * 16 + tid], a1);
      }
      sXU[tid] = fmaxf(a0, 0.f);
      sXV[tid] = fmaxf(a1, 0.f);  // XU_joint (pre-product)
    }
    __syncthreads();

    float lg = -3.0e38f;
    if (tid < 200) {
      float s = 0.f;
      const float* yv = &sYV[tid * 33];
#pragma unroll
      for (int k = 0; k < 16; ++k) s = fmaf(sXU[k] * yv[k], p_g[g * 16 + k], s);
      lg = s;
    }
    sRed[tid] = lg;
    __syncthreads();
    for (int st = 128; st > 0; st >>= 1) {
      if (tid < st) sRed[tid] = fmaxf(sRed[tid], sRed[tid + st]);
      __syncthreads();
    }
    float mx = sRed[0];
    __syncthreads();
    float e = (tid < 200) ? expf(lg - mx) : 0.f;
    sRed[tid] = e;
    __syncthreads();
    for (int st = 128; st > 0; st >>= 1) {
      if (tid < st) sRed[tid] += sRed[tid + st];
      __syncthreads();
    }
    float inv = 1.0f / sRed[0];
    __syncthreads();
    float at = e * inv;
    if (tid < 200) {
      sVec[tid] = at;
      ((g == 0) ? attn0 : attn1)[(size_t)b * 200 + tid] = at;
    }
    __syncthreads();

    // Vy_hat[k] = sum_m attn[m]*YVj[m,k]; fold product with XU_joint
    if (tid < 16) {
      float s = 0.f;
      for (int m = 0; m < 200; ++m) s = fmaf(sVec[m], sYV[m * 33 + 16 + tid], s);
      sVy[tid] = s * sXV[tid];
    }
    __syncthreads();

    float fsum = 0.f;
    if (tid < 20) {
      float fj = 0.f;
      for (int k = 0; k < 16; ++k) fj = fmaf(sVy[k], P[k * 20 + tid], fj);
      fsum = sF[tid] + fj;
      sRed[tid] = fsum;
    }
    __syncthreads();
    if (tid == 0) {
      float mu = 0.f;
      for (int j = 0; j < 20; ++j) mu += sRed[j];
      mu *= (1.f / 20.f);
      float var = 0.f;
      for (int j = 0; j < 20; ++j) { float d0 = sRed[j] - mu; var = fmaf(d0, d0, var); }
      var *= (1.f / 20.f);
      sLN[0] = mu;
      sLN[1] = 1.0f / sqrtf(var + 1e-5f);
    }
    __syncthreads();
    if (tid < 20)
      sF[tid] = fmaf((fsum - sLN[0]) * sLN[1], res_g[g * 20 + tid], res_b[g * 20 + tid]);
    __syncthreads();
  }

  // FC head: 20 -> 256 (LN, relu) -> 64 (LN, relu) -> 1
  float o1 = fc1_b[tid];
  for (int j = 0; j < 20; ++j) o1 = fmaf(sF[j], fc1_w[j * 256 + tid], o1);
  sRed[tid] = o1;
  __syncthreads();
  for (int st = 128; st > 0; st >>= 1) {
    if (tid < st) sRed[tid] += sRed[tid + st];
    __syncthreads();
  }
  float mu1 = sRed[0] * (1.f / 256.f);
  __syncthreads();
  float d1 = o1 - mu1;
  sRed[tid] = d1 * d1;
  __syncthreads();
  for (int st = 128; st > 0; st >>= 1) {
    if (tid < st) sRed[tid] += sRed[tid + st];
    __syncthreads();
  }
  float rs1 = 1.0f / sqrtf(sRed[0] * (1.f / 256.f) + 1e-5f);
  __syncthreads();
  sO1[tid] = fmaxf(fmaf(d1 * rs1, bn1_g[tid], bn1_b[tid]), 0.f);
  __syncthreads();

  float o2 = 0.f;
  if (tid < 64) {
    o2 = fc2_b[tid];
    for (int k = 0; k < 256; ++k) o2 = fmaf(sO1[k], fc2_w[k * 64 + tid], o2);
  }
  sRed[tid] = (tid < 64) ? o2 : 0.f;
  __syncthreads();
  for (int st = 128; st > 0; st >>= 1) {
    if (tid < st) sRed[tid] += sRed[tid + st];
    __syncthreads();
  }
  float mu2 = sRed[0] * (1.f / 64.f);
  __syncthreads();
  float d2 = o2 - mu2;
  sRed[tid] = (tid < 64) ? d2 * d2 : 0.f;
  __syncthreads();
  for (int st = 128; st > 0; st >>= 1) {
    if (tid < st) sRed[tid] += sRed[tid + st];
    __syncthreads();
  }
  float rs2 = 1.0f / sqrtf(sRed[0] * (1.f / 64.f) + 1e-5f);
  __syncthreads();
  float h2v = 0.f;
  if (tid < 64)
    h2v = fmaxf(fmaf(d2 * rs2, bn2_g[tid], bn2_b[tid]), 0.f) * fc3_w[tid];
  sRed[tid] = h2v;
  __syncthreads();
  for (int st = 128; st > 0; st >>= 1) {
    if (tid < st) sRed[tid] += sRed[tid + st];
    __syncthreads();
  }
  if (tid == 0) out[b] = sRed[0] + fc3_b[0];
}

// ---------------------------------------------------------------------------
extern "C" void kernel_launch(void* const* d_in, const int* in_sizes, int n_in,
                              void* d_out, int out_size, void* d_ws, size_t ws_size,
                              hipStream_t stream) {
  (void)in_sizes; (void)n_in; (void)out_size; (void)ws_size;
  const float* adj     = (const float*)d_in[0];
  const float* feat    = (const float*)d_in[1];
  const float* desc    = (const float*)d_in[2];
  const float* W_gc1   = (const float*)d_in[3];
  const float* W_gc2   = (const float*)d_in[4];
  const float* id_emb  = (const float*)d_in[5];
  const float* val_w1  = (const float*)d_in[6];
  const float* val_b1  = (const float*)d_in[7];
  const float* val_w2  = (const float*)d_in[8];
  const float* val_b2  = (const float*)d_in[9];
  const float* tok_g   = (const float*)d_in[10];
  const float* tok_b   = (const float*)d_in[11];
  const float* U_attn  = (const float*)d_in[12];
  const float* V_attn  = (const float*)d_in[13];
  const float* p_g     = (const float*)d_in[14];
  const float* U_joint = (const float*)d_in[15];
  const float* V_joint = (const float*)d_in[16];
  const float* P       = (const float*)d_in[17];
  const float* res_g   = (const float*)d_in[18];
  const float* res_b   = (const float*)d_in[19];
  const float* fc1_w   = (const float*)d_in[20];
  const float* fc1_b   = (const float*)d_in[21];
  const float* fc2_w   = (const float*)d_in[22];
  const float* fc2_b   = (const float*)d_in[23];
  const float* fc3_w   = (const float*)d_in[24];
  const float* fc3_b   = (const float*)d_in[25];
  const float* bn1_g   = (const float*)d_in[26];
  const float* bn1_b   = (const float*)d_in[27];
  const float* bn2_g   = (const float*)d_in[28];
  const float* bn2_b   = (const float*)d_in[29];

  const int B = 4096;
  float* hgw = (float*)d_ws;           // B*20 floats of scratch
  float* outp = (float*)d_out;         // (B,1)
  float* attn0 = outp + B;             // (B,200)
  float* attn1 = attn0 + (size_t)B * 200;

  const size_t shA = (size_t)(64 * 65 + 64 * 116 + 64 * 116 + 64 * 116 +
                              100 * 33 + 64 + 64 * 20) * sizeof(float);
  gcn_kernel<<<B, 256, shA, stream>>>(adj, feat, W_gc1, W_gc2, hgw);

  const size_t shB = (size_t)(208 * 33 + 208 * 65 + 2048 + 2048 + 208 * 33 +
                              256 + 256 + 256 + 20 + 16 + 16 + 16 + 8) *
                     sizeof(float);
  attn_head_kernel<<<B, 256, shB, stream>>>(
      desc, id_emb, val_w1, val_b1, val_w2, val_b2, tok_g, tok_b, U_attn,
      V_attn, p_g, U_joint, V_joint, P, res_g, res_b, fc1_w, fc1_b, fc2_w,
      fc2_b, fc3_w, fc3_b, bn1_g, bn1_b, bn2_g, bn2_b, hgw, outp, attn0, attn1);
}